// model_52089363366197
// MI455X (gfx1250) — compile-verified
//
#include <hip/hip_runtime.h>
#include <math.h>

typedef __attribute__((ext_vector_type(2))) float v2f;
typedef __attribute__((ext_vector_type(8))) float v8f;

#define NEG_SLOPE 0.2f

__device__ __forceinline__ float lrelu(float v) { return v >= 0.f ? v : NEG_SLOPE * v; }

// Float atomic max via sign-split integer atomics (exact for IEEE754, works with -inf init).
__device__ __forceinline__ void atomicMaxF(float* addr, float val) {
  if (val >= 0.f) atomicMax((int*)addr, __float_as_int(val));
  else            atomicMin((unsigned int*)addr, __float_as_uint(val));
}

// ---------------------------------------------------------------------------
// GEMM1: H[100000x64] = X[100000x256] @ W[256x64], fp32 WMMA 16x16x4.
// grid.x = N/16 (M tiles), block = 128 (4 waves), wave w owns N-columns [16w,16w+16).
// A layout: lane l -> row M=l%16, K = 2*(l/16)+{0,1} (contiguous float2 from row-major X).
// B layout: lane l -> col N=l%16, K = 2*(l/16)+{0,1}.
// D layout: VGPR v -> M = v + (lane>=16 ? 8 : 0), N = l%16.
// ---------------------------------------------------------------------------
__global__ void gemm1_x_w1(const float* __restrict__ X, const float* __restrict__ W,
                           float* __restrict__ H) {
  const int lane = threadIdx.x & 31;
  const int wave = threadIdx.x >> 5;
  const int kgrp = (lane >> 4) << 1;               // 0 or 2
  const int ncol = (wave << 4) + (lane & 15);      // 0..63
  const float* xrow = X + ((long)blockIdx.x * 16 + (lane & 15)) * 256;
  v8f acc = {};
#pragma unroll 8
  for (int k0 = 0; k0 < 256; k0 += 4) {
    v2f a = *(const v2f*)(xrow + k0 + kgrp);
    v2f b;
    b.x = W[(k0 + kgrp) * 64 + ncol];
    b.y = W[(k0 + kgrp + 1) * 64 + ncol];
    acc = __builtin_amdgcn_wmma_f32_16x16x4_f32(false, a, false, b, (short)0, acc,
                                                false, false);
  }
  const long mb = (long)blockIdx.x * 16 + ((lane >> 4) << 3);
#pragma unroll
  for (int v = 0; v < 8; ++v)
    H[(mb + v) * 64 + ncol] = acc[v];
}

// ---------------------------------------------------------------------------
// GEMM2: H2[100000x8] = relu(A1)[100000x64] @ W[64x8] (N padded to 16 in-register).
// block = 128 (4 waves), each wave owns a distinct M-tile.
// ---------------------------------------------------------------------------
__global__ void gemm2_h_w2(const float* __restrict__ A1, const float* __restrict__ W,
                           float* __restrict__ H2, int mtiles) {
  const int lane = threadIdx.x & 31;
  const int wave = threadIdx.x >> 5;
  const int mtile = blockIdx.x * 4 + wave;
  if (mtile >= mtiles) return;                     // wave-uniform: EXEC stays all-1s
  const int kgrp = (lane >> 4) << 1;
  const int ncol = lane & 15;
  const float* arow = A1 + ((long)mtile * 16 + ncol) * 64;
  v8f acc = {};
#pragma unroll
  for (int k0 = 0; k0 < 64; k0 += 4) {
    v2f a = *(const v2f*)(arow + k0 + kgrp);
    a.x = fmaxf(a.x, 0.f);                         // fused ReLU of layer-1 output
    a.y = fmaxf(a.y, 0.f);
    v2f b;                                          // cols 8..15 read col&7 (garbage, never stored)
    b.x = W[(k0 + kgrp) * 8 + (ncol & 7)];
    b.y = W[(k0 + kgrp + 1) * 8 + (ncol & 7)];
    acc = __builtin_amdgcn_wmma_f32_16x16x4_f32(false, a, false, b, (short)0, acc,
                                                false, false);
  }
  if (ncol < 8) {
    const long mb = (long)mtile * 16 + ((lane >> 4) << 3);
#pragma unroll
    for (int v = 0; v < 8; ++v)
      H2[(mb + v) * 8 + ncol] = acc[v];
  }
}

// al_src/al_dst logits: one thread per (node, head); hf is [N,H,C] contiguous.
template <int H, int C>
__global__ void attn_logits(const float* __restrict__ hf, const float* __restrict__ as,
                            const float* __restrict__ ad, float* __restrict__ als,
                            float* __restrict__ ald, long NH) {
  long t = (long)blockIdx.x * blockDim.x + threadIdx.x;
  if (t >= NH) return;
  const int h = (int)(t % H);
  const float* hp = hf + t * C;
  float ss = 0.f, sd = 0.f;
#pragma unroll
  for (int c = 0; c < C; ++c) {
    float v = hp[c];
    ss += v * as[h * C + c];
    sd += v * ad[h * C + c];
  }
  als[t] = ss;
  ald[t] = sd;
}

__global__ void init_softmax(float* __restrict__ m, float* __restrict__ den, long n) {
  long t = (long)blockIdx.x * blockDim.x + threadIdx.x;
  if (t < n) { m[t] = -__builtin_huge_valf(); den[t] = 0.f; }
}

__global__ void init_acc_bias(float* __restrict__ acc, const float* __restrict__ b,
                              long n, int F) {
  long t = (long)blockIdx.x * blockDim.x + threadIdx.x;
  if (t < n) acc[t] = b[(int)(t % F)];
}

__device__ __forceinline__ void edge_sd(const int* __restrict__ ei, long Eraw, long e,
                                        int& s, int& d) {
  if (e < Eraw) { s = ei[e]; d = ei[Eraw + e]; }
  else          { s = d = (int)(e - Eraw); }       // implicit self-loops
}

// Pass 1: per-destination segment max of leaky_relu(al_src[s]+al_dst[d]).
template <int H>
__global__ void edge_softmax_max(const int* __restrict__ ei, long Eraw, long Etot,
                                 const float* __restrict__ als, const float* __restrict__ ald,
                                 float* __restrict__ m) {
  long e = (long)blockIdx.x * blockDim.x + threadIdx.x;
  if (e >= Etot) return;
  int s, d; edge_sd(ei, Eraw, e, s, d);
#pragma unroll
  for (int h = 0; h < H; ++h) {
    float v = lrelu(als[(long)s * H + h] + ald[(long)d * H + h]);
    atomicMaxF(&m[(long)d * H + h], v);
  }
}

// Pass 2: per-destination sum of exp(e - m[dst]).
template <int H>
__global__ void edge_softmax_sum(const int* __restrict__ ei, long Eraw, long Etot,
                                 const float* __restrict__ als, const float* __restrict__ ald,
                                 const float* __restrict__ m, float* __restrict__ den) {
  long e = (long)blockIdx.x * blockDim.x + threadIdx.x;
  if (e >= Etot) return;
  int s, d; edge_sd(ei, Eraw, e, s, d);
#pragma unroll
  for (int h = 0; h < H; ++h) {
    float v = lrelu(als[(long)s * H + h] + ald[(long)d * H + h]);
    atomicAdd(&den[(long)d * H + h], __expf(v - m[(long)d * H + h]));
  }
}

// Pass 3: out[dst] += alpha * h[src]; one thread per (edge, head), C atomics each.
template <int H, int C>
__global__ void edge_aggregate(const int* __restrict__ ei, long Eraw, long Etot,
                               const float* __restrict__ als, const float* __restrict__ ald,
                               const float* __restrict__ m, const float* __restrict__ den,
                               const float* __restrict__ hf, float* __restrict__ acc) {
  long t = (long)blockIdx.x * blockDim.x + threadIdx.x;
  if (t >= Etot * H) return;
  long e = t / H;
  int h = (int)(t % H);
  int s, d; edge_sd(ei, Eraw, e, s, d);
  long sh = (long)s * H + h, dh = (long)d * H + h;
  float v = lrelu(als[sh] + ald[dh]);
  float alpha = __expf(v - m[dh]) / (den[dh] + 1e-16f);
  const float* hp = hf + sh * C;
  float* op = acc + dh * C;
#pragma unroll
  for (int c = 0; c < C; ++c)
    atomicAdd(&op[c], hp[c] * alpha);
}

__global__ void score_kernel(const float* __restrict__ emb, const float* __restrict__ W3,
                             const float* __restrict__ b3, float* __restrict__ score, long N) {
  long n = (long)blockIdx.x * blockDim.x + threadIdx.x;
  if (n >= N) return;
  float s = b3[0];
#pragma unroll
  for (int c = 0; c < 8; ++c) s += emb[n * 8 + c] * W3[c];
  score[n] = s;
}

static inline int nblk(long n, int b) { return (int)((n + b - 1) / b); }

extern "C" void kernel_launch(void* const* d_in, const int* in_sizes, int n_in,
                              void* d_out, int out_size, void* d_ws, size_t ws_size,
                              hipStream_t stream) {
  const float* x   = (const float*)d_in[0];
  const int*   ei  = (const int*)d_in[1];
  const float* W1  = (const float*)d_in[2];
  const float* as1 = (const float*)d_in[3];
  const float* ad1 = (const float*)d_in[4];
  const float* b1  = (const float*)d_in[5];
  const float* W2  = (const float*)d_in[6];
  const float* as2 = (const float*)d_in[7];
  const float* ad2 = (const float*)d_in[8];
  const float* b2  = (const float*)d_in[9];
  const float* W3  = (const float*)d_in[10];
  const float* b3  = (const float*)d_in[11];

  const long N    = in_sizes[0] / 256;   // 100000
  const long Eraw = in_sizes[1] / 2;     // 3,200,000
  const long Etot = Eraw + N;            // edges + self-loops

  float* ws   = (float*)d_ws;
  float* h1   = ws;                      // [N,64]
  float* acc1 = h1 + N * 64;             // [N,64] (bias pre-loaded, ReLU applied in GEMM2)
  float* als1 = acc1 + N * 64;           // [N,8]
  float* ald1 = als1 + N * 8;            // [N,8]
  float* m1   = ald1 + N * 8;            // [N,8]
  float* den1 = m1 + N * 8;              // [N,8]
  float* h2   = den1 + N * 8;            // [N,8]
  float* als2 = h2 + N * 8;              // [N]
  float* ald2 = als2 + N;                // [N]
  float* m2   = ald2 + N;                // [N]
  float* den2 = m2 + N;                  // [N]

  float* emb   = (float*)d_out;          // [N,8]
  float* score = emb + N * 8;            // [N,1]

  const int B = 256;
  const int mtiles = (int)(N / 16);      // 6250

  // ---- Layer 1 ----
  gemm1_x_w1<<<mtiles, 128, 0, stream>>>(x, W1, h1);
  attn_logits<8, 8><<<nblk(N * 8, B), B, 0, stream>>>(h1, as1, ad1, als1, ald1, N * 8);
  init_softmax<<<nblk(N * 8, B), B, 0, stream>>>(m1, den1, N * 8);
  init_acc_bias<<<nblk(N * 64, B), B, 0, stream>>>(acc1, b1, N * 64, 64);
  edge_softmax_max<8><<<nblk(Etot, B), B, 0, stream>>>(ei, Eraw, Etot, als1, ald1, m1);
  edge_softmax_sum<8><<<nblk(Etot, B), B, 0, stream>>>(ei, Eraw, Etot, als1, ald1, m1, den1);
  edge_aggregate<8, 8><<<nblk(Etot * 8, B), B, 0, stream>>>(ei, Eraw, Etot, als1, ald1,
                                                            m1, den1, h1, acc1);

  // ---- Layer 2 (A = relu(acc1) fused into GEMM2 load) ----
  gemm2_h_w2<<<(mtiles + 3) / 4, 128, 0, stream>>>(acc1, W2, h2, mtiles);
  attn_logits<1, 8><<<nblk(N, B), B, 0, stream>>>(h2, as2, ad2, als2, ald2, N);
  init_softmax<<<nblk(N, B), B, 0, stream>>>(m2, den2, N);
  init_acc_bias<<<nblk(N * 8, B), B, 0, stream>>>(emb, b2, N * 8, 8);
  edge_softmax_max<1><<<nblk(Etot, B), B, 0, stream>>>(ei, Eraw, Etot, als2, ald2, m2);
  edge_softmax_sum<1><<<nblk(Etot, B), B, 0, stream>>>(ei, Eraw, Etot, als2, ald2, m2, den2);
  edge_aggregate<1, 8><<<nblk(Etot, B), B, 0, stream>>>(ei, Eraw, Etot, als2, ald2,
                                                        m2, den2, h2, emb);

  // ---- Score head ----
  score_kernel<<<nblk(N, B), B, 0, stream>>>(emb, W3, b3, score, N);
}